// EfficientMemorySoftmax_73907797230271
// MI455X (gfx1250) — compile-verified
//
#include <hip/hip_runtime.h>
#include <hip/hip_bf16.h>

#define AS1 __attribute__((address_space(1)))
#define AS3 __attribute__((address_space(3)))

typedef int v4i __attribute__((vector_size(4 * sizeof(int))));

#if defined(__has_builtin)
#  if __has_builtin(__builtin_amdgcn_global_load_async_to_lds_b128)
#    define HAVE_ASYNC_LDS 1
#  else
#    define HAVE_ASYNC_LDS 0
#  endif
#  if __has_builtin(__builtin_amdgcn_s_wait_asynccnt)
#    define HAVE_WAIT_ASYNC 1
#  else
#    define HAVE_WAIT_ASYNC 0
#  endif
#else
#  define HAVE_ASYNC_LDS 0
#  define HAVE_WAIT_ASYNC 0
#endif

__device__ __forceinline__ float wave_reduce_max(float v) {
#pragma unroll
  for (int off = 16; off > 0; off >>= 1)
    v = fmaxf(v, __shfl_xor(v, off, 32));
  return v;
}

__device__ __forceinline__ float wave_reduce_sum(float v) {
#pragma unroll
  for (int off = 16; off > 0; off >>= 1)
    v += __shfl_xor(v, off, 32);
  return v;
}

// One 256-thread block (8 wave32) per row of 2048 fp32 values.
// Row is staged global->LDS with the gfx1250 async copy engine (ASYNCcnt),
// then processed entirely from registers: 1 global read + 1 global write
// per element (bandwidth floor ~1 GB @ 23.3 TB/s).
__global__ __launch_bounds__(256) void EfficientMemorySoftmax_73907797230271_kernel(
    const float* __restrict__ x, float* __restrict__ out, long long n_rows) {
  constexpr int ROW = 2048;
  const long long row = blockIdx.x;
  if (row >= n_rows) return;

  const float* __restrict__ src = x + row * (long long)ROW;
  float* __restrict__ dst = out + row * (long long)ROW;
  const int tid = threadIdx.x;

  __shared__ float stage[ROW];  // 8 KB staging buffer for the async copy
  __shared__ float red[16];     // cross-wave reduction scratch (8 max + 8 sum)

  float4 a, b;

#if HAVE_ASYNC_LDS
  {
    // Each lane copies its own two 16B chunks: [tid*4, tid*4+4) and +1024.
    AS1 v4i* g0 = (AS1 v4i*)(src + tid * 4);
    AS1 v4i* g1 = (AS1 v4i*)(src + 1024 + tid * 4);
    AS3 v4i* l0 = (AS3 v4i*)(stage + tid * 4);
    AS3 v4i* l1 = (AS3 v4i*)(stage + 1024 + tid * 4);
    __builtin_amdgcn_global_load_async_to_lds_b128(g0, l0, /*offset=*/0, /*cpol=*/0);
    __builtin_amdgcn_global_load_async_to_lds_b128(g1, l1, /*offset=*/0, /*cpol=*/0);
  }
#if HAVE_WAIT_ASYNC
  __builtin_amdgcn_s_wait_asynccnt(0);
#else
  asm volatile("s_wait_asynccnt 0" ::: "memory");
#endif
  asm volatile("" ::: "memory");  // keep the ds_loads after the wait
  // Each lane reads back exactly the LDS bytes it staged itself -> no barrier.
  a = *reinterpret_cast<const float4*>(stage + tid * 4);
  b = *reinterpret_cast<const float4*>(stage + 1024 + tid * 4);
#else
  a = *reinterpret_cast<const float4*>(src + tid * 4);
  b = *reinterpret_cast<const float4*>(src + 1024 + tid * 4);
#endif

  // ---- row max ----
  float m = fmaxf(fmaxf(fmaxf(a.x, a.y), fmaxf(a.z, a.w)),
                  fmaxf(fmaxf(b.x, b.y), fmaxf(b.z, b.w)));
  m = wave_reduce_max(m);
  const int wave = tid >> 5;
  if ((tid & 31) == 0) red[wave] = m;
  __syncthreads();
  float rowmax = red[0];
#pragma unroll
  for (int i = 1; i < 8; ++i) rowmax = fmaxf(rowmax, red[i]);

  // ---- exp + row sum ----
  a.x = __expf(a.x - rowmax);
  a.y = __expf(a.y - rowmax);
  a.z = __expf(a.z - rowmax);
  a.w = __expf(a.w - rowmax);
  b.x = __expf(b.x - rowmax);
  b.y = __expf(b.y - rowmax);
  b.z = __expf(b.z - rowmax);
  b.w = __expf(b.w - rowmax);
  float s = (a.x + a.y) + (a.z + a.w) + (b.x + b.y) + (b.z + b.w);
  s = wave_reduce_sum(s);
  if ((tid & 31) == 0) red[8 + wave] = s;
  __syncthreads();
  float tot = 0.0f;
#pragma unroll
  for (int i = 0; i < 8; ++i) tot += red[8 + i];

  const float inv = __builtin_amdgcn_rcpf(tot);

  a.x *= inv; a.y *= inv; a.z *= inv; a.w *= inv;
  b.x *= inv; b.y *= inv; b.z *= inv; b.w *= inv;

  *reinterpret_cast<float4*>(dst + tid * 4) = a;
  *reinterpret_cast<float4*>(dst + 1024 + tid * 4) = b;
}

extern "C" void kernel_launch(void* const* d_in, const int* in_sizes, int n_in,
                              void* d_out, int out_size, void* d_ws, size_t ws_size,
                              hipStream_t stream) {
  (void)n_in; (void)d_ws; (void)ws_size; (void)out_size;
  const float* x = (const float*)d_in[0];
  float* out = (float*)d_out;
  const long long total = (long long)in_sizes[0];
  const long long rows = total / 2048;  // (2*16*2048) = 65536 rows of 2048
  dim3 grid((unsigned)rows);
  dim3 block(256);
  EfficientMemorySoftmax_73907797230271_kernel<<<grid, block, 0, stream>>>(x, out, rows);
}